// GCMCGraphConv_14456859918894
// MI455X (gfx1250) — compile-verified
//
#include <hip/hip_runtime.h>

typedef __attribute__((ext_vector_type(2))) float v2f;
typedef __attribute__((ext_vector_type(8))) float v8f;

#define OUTF 64
#define TILE_M 16
#define WAVES_PER_BLOCK 8

__global__ void __launch_bounds__(256)
gcmc_zero_kernel(float* __restrict__ out, int n4) {
    int i = blockIdx.x * blockDim.x + threadIdx.x;
    if (i < n4) {
        ((float4*)out)[i] = make_float4(0.f, 0.f, 0.f, 0.f);
    }
}

__global__ void __launch_bounds__(256)
gcmc_edge_kernel(const float* __restrict__ weight,       // [N_SRC, 64]
                 const float* __restrict__ prob_w,       // [64]
                 const float* __restrict__ score_w,      // [64]
                 const float* __restrict__ review_w,     // [64, 64]
                 const float* __restrict__ review_feat,  // [E, 64]
                 const float* __restrict__ cj,           // [N_SRC]
                 const float* __restrict__ ci,           // [N_DST]
                 const int*   __restrict__ src_idx,      // [E]
                 const int*   __restrict__ dst_idx,      // [E]
                 float*       __restrict__ out,          // [N_DST, 64]
                 int ntiles)
{
    const int lane = threadIdx.x & 31;
    const int wid  = threadIdx.x >> 5;
    const int tile = blockIdx.x * WAVES_PER_BLOCK + wid;
    if (tile >= ntiles) return;              // wave-uniform: EXEC stays all-1s for WMMA

    const int  m  = lane & 15;               // edge-in-tile owned by this lane
    const int  hi = lane >> 4;               // wave half
    const long e  = (long)tile * TILE_M + m; // edge id (duplicated across halves)

    // ---------------- Stage 1: per-edge gate scalars ----------------
    // Halves split K: lanes 0-15 do k=[0,32), lanes 16-31 do k=[32,64).
    const float4* rrow4 = (const float4*)(review_feat + e * OUTF) + hi * 8;
    const float4* pw4   = (const float4*)prob_w  + hi * 8;
    const float4* sw4   = (const float4*)score_w + hi * 8;
    float accp = 0.f, accs = 0.f;
#pragma unroll
    for (int i = 0; i < 8; ++i) {
        float4 v = rrow4[i];
        float4 p = pw4[i];
        float4 s = sw4[i];
        accp += v.x * p.x + v.y * p.y + v.z * p.z + v.w * p.w;
        accs += v.x * s.x + v.y * s.y + v.z * s.z + v.w * s.w;
    }
    accp += __shfl_xor(accp, 16, 32);
    accs += __shfl_xor(accs, 16, 32);
    float pa = 1.f / (1.f + __expf(-accp));   // sigmoid(review_feat @ prob_w)
    float sg = 1.f / (1.f + __expf(-accs));   // sigmoid(review_feat @ review_score_w)

    const int   si = src_idx[e];
    const int   di = dst_idx[e];
    // ci[dst] distributes over the segment sum -> fold it into the edge scale.
    const float cc = cj[si] * ci[di];

    // Broadcast the 8 per-edge scalar sets this half needs (C row M = r + hi*8).
    float pa_r[8], sg_r[8], cc_r[8];
    int   si_r[8], di_r[8];
#pragma unroll
    for (int r = 0; r < 8; ++r) {
        int q = r + hi * 8;                   // lane that owns edge (r + hi*8)
        pa_r[r] = __shfl(pa, q, 32);
        sg_r[r] = __shfl(sg, q, 32);
        cc_r[r] = __shfl(cc, q, 32);
        si_r[r] = __shfl(si, q, 32);
        di_r[r] = __shfl(di, q, 32);
    }

    // ---------------- Stage 2: 16x64 @ 64x64 fp32 WMMA GEMM ----------------
    // C[mT][n] = sum_k A[mT][k] * B[k][n],  A = review_feat tile, B[k][n] = review_w[n][k]
    // A 16x4 f32 layout: lanes 0-15 hold K=k0+0,k0+1; lanes 16-31 hold K=k0+2,k0+3.
    v8f c0 = {}, c1 = {}, c2 = {}, c3 = {};
    const float* arow = review_feat + ((long)tile * TILE_M + m) * OUTF + hi * 2;
    const float* b0p  = review_w + (long)(m +  0) * OUTF + hi * 2;
    const float* b1p  = review_w + (long)(m + 16) * OUTF + hi * 2;
    const float* b2p  = review_w + (long)(m + 32) * OUTF + hi * 2;
    const float* b3p  = review_w + (long)(m + 48) * OUTF + hi * 2;
#pragma unroll
    for (int k0 = 0; k0 < OUTF; k0 += 4) {
        v2f a  = *(const v2f*)(arow + k0);
        v2f b0 = *(const v2f*)(b0p + k0);
        v2f b1 = *(const v2f*)(b1p + k0);
        v2f b2 = *(const v2f*)(b2p + k0);
        v2f b3 = *(const v2f*)(b3p + k0);
        c0 = __builtin_amdgcn_wmma_f32_16x16x4_f32(false, a, false, b0, (short)0, c0, false, false);
        c1 = __builtin_amdgcn_wmma_f32_16x16x4_f32(false, a, false, b1, (short)0, c1, false, false);
        c2 = __builtin_amdgcn_wmma_f32_16x16x4_f32(false, a, false, b2, (short)0, c2, false, false);
        c3 = __builtin_amdgcn_wmma_f32_16x16x4_f32(false, a, false, b3, (short)0, c3, false, false);
    }

    // ---------------- Stage 3: gate, gather weight, scatter-add ----------------
    // C VGPR r, lane l: edge mT = r + hi*8, out col n = n0 + (l & 15).
#pragma unroll
    for (int t = 0; t < 4; ++t) {
        v8f cacc = (t == 0) ? c0 : (t == 1) ? c1 : (t == 2) ? c2 : c3;
        const int ncol = t * 16 + m;
#pragma unroll
        for (int r = 0; r < 8; ++r) {
            float w   = weight[(long)si_r[r] * OUTF + ncol];        // L2-resident gather
            float val = (w * pa_r[r] + cacc[r] * sg_r[r]) * cc_r[r];
            atomicAdd(out + (long)di_r[r] * OUTF + ncol, val);      // global_atomic_add_f32
        }
    }
}

extern "C" void kernel_launch(void* const* d_in, const int* in_sizes, int n_in,
                              void* d_out, int out_size, void* d_ws, size_t ws_size,
                              hipStream_t stream) {
    const float* weight      = (const float*)d_in[0];
    const float* prob_w      = (const float*)d_in[1];
    const float* score_w     = (const float*)d_in[2];
    const float* review_w    = (const float*)d_in[3];
    const float* review_feat = (const float*)d_in[4];
    const float* cj          = (const float*)d_in[5];
    const float* ci          = (const float*)d_in[6];
    const int*   src_idx     = (const int*)d_in[7];
    const int*   dst_idx     = (const int*)d_in[8];
    float*       out         = (float*)d_out;

    const int E = in_sizes[7];           // 1,600,000
    const int n4 = out_size / 4;         // out is N_DST*64 floats, divisible by 4
    gcmc_zero_kernel<<<(n4 + 255) / 256, 256, 0, stream>>>(out, n4);

    const int ntiles = E / TILE_M;       // E is a multiple of 16
    const int blocks = (ntiles + WAVES_PER_BLOCK - 1) / WAVES_PER_BLOCK;
    gcmc_edge_kernel<<<blocks, 256, 0, stream>>>(weight, prob_w, score_w, review_w,
                                                 review_feat, cj, ci, src_idx, dst_idx,
                                                 out, ntiles);
}